// KNN_68152541053158
// MI455X (gfx1250) — compile-verified
//
#include <hip/hip_runtime.h>

typedef __attribute__((ext_vector_type(2))) float v2f;
typedef __attribute__((ext_vector_type(8))) float v8f;

#define NPTS    8192
#define KSEL    16
#define CHUNK   2048     // dst points staged in LDS per pass (32KB)
#define THREADS 256
#define QPB     128      // queries per block (8 waves * 16)

struct __align__(16) Pt4 { float x, y, z, w; };

__device__ __forceinline__ void insert8(const v8f& dmat, int midx,
                                        float (&l)[KSEL], int (&li)[KSEL])
{
#pragma unroll
    for (int r = 0; r < 8; ++r) {
        const float v  = dmat[r];
        const int   id = midx + r;
        if (v < l[KSEL - 1]) {
            bool done = false;
#pragma unroll
            for (int j = KSEL - 1; j > 0; --j) {
                const bool shift = (!done) & (v < l[j - 1]);
                const float pv = done ? l[j]  : (shift ? l[j - 1]  : v);
                const int   pi = done ? li[j] : (shift ? li[j - 1] : id);
                done = done | (!shift);
                l[j]  = pv;
                li[j] = pi;
            }
            if (!done) { l[0] = v; li[0] = id; }
        }
    }
}

__global__ __launch_bounds__(THREADS)
void KNN_68152541053158_kernel(const float* __restrict__ src,
                               const float* __restrict__ dst,
                               float* __restrict__ outD,
                               int*   __restrict__ outI)
{
    // 32KB LDS: candidate stage; reused at the end as merge scratch (16KB+16KB).
    __shared__ Pt4 cand[CHUNK];
    float* lval = (float*)cand;                   // 256*16 floats = 16KB
    int*   lidx = (int*)cand + THREADS * KSEL;    // 256*16 ints   = 16KB

    const int tid    = threadIdx.x;
    const int lane   = tid & 31;
    const int wave   = tid >> 5;
    const int batch  = blockIdx.x >> 6;           // 64 blocks per batch
    const int qbase  = (blockIdx.x & 63) * QPB + wave * 16;
    const int q      = qbase + (lane & 15);       // this lane's query (src point)
    const bool hi    = lane >= 16;
    const int lane15 = lane & 15;

    // ---- load query point; build per-lane B column + C splat ----
    const float* sp = src + ((size_t)batch * NPTS + q) * 3;
    const float sx = sp[0], sy = sp[1], sz = sp[2];
    const float qs = sx * sx + sy * sy + sz * sz;

    // B (4x16, KxN): VGPR0 holds rows K0|K2 across lane halves, VGPR1 = K1|K3
    v2f bmat;
    bmat.x = hi ? (-2.0f * sz) : (-2.0f * sx);
    bmat.y = hi ? 1.0f         : (-2.0f * sy);

    v8f cmat;
#pragma unroll
    for (int r = 0; r < 8; ++r) cmat[r] = qs;     // C[m][n] = ||s_n||^2 (per-lane splat)

    // ---- per-lane top-16 (sorted ascending on squared distance) ----
    float l[KSEL];
    int   li[KSEL];
#pragma unroll
    for (int j = 0; j < KSEL; ++j) { l[j] = 3.0e38f; li[j] = 0; }

    const float* dbase = dst + (size_t)batch * NPTS * 3;

    for (int chunk = 0; chunk < NPTS; chunk += CHUNK) {
        __syncthreads();
        // cooperative stage: [x, y, z, ||d||^2] per candidate
        for (int i = tid; i < CHUNK; i += THREADS) {
            const float* dp = dbase + (size_t)(chunk + i) * 3;
            const float x = dp[0], y = dp[1], z = dp[2];
            Pt4 p; p.x = x; p.y = y; p.z = z; p.w = x * x + y * y + z * z;
            cand[i] = p;
        }
        __syncthreads();

        // software-pipelined scan: 32 candidates (2 WMMA tiles) per iteration
        Pt4 pa = cand[lane15];
        Pt4 pb = cand[16 + lane15];
        for (int tile = 0; tile < CHUNK; tile += 32) {
            v2f a0, a1;
            a0.x = hi ? pa.z : pa.x;  a0.y = hi ? pa.w : pa.y;
            a1.x = hi ? pb.z : pb.x;  a1.y = hi ? pb.w : pb.y;

            // prefetch next iteration's tiles (uniform clamp, no divergence)
            const int nt = (tile + 32 < CHUNK) ? (tile + 32) : 0;
            pa = cand[nt + lane15];
            pb = cand[nt + 16 + lane15];

            const v8f d0 = __builtin_amdgcn_wmma_f32_16x16x4_f32(
                false, a0, false, bmat, (short)0, cmat, false, false);
            const v8f d1 = __builtin_amdgcn_wmma_f32_16x16x4_f32(
                false, a1, false, bmat, (short)0, cmat, false, false);

            // one cheap reject test for all 16 new candidates of this lane
            float mn = fminf(d0[0], d1[0]);
#pragma unroll
            for (int r = 1; r < 8; ++r) mn = fminf(mn, fminf(d0[r], d1[r]));

            if (mn < l[KSEL - 1]) {
                const int base = chunk + tile + (hi ? 8 : 0);
                insert8(d0, base,      l, li);
                insert8(d1, base + 16, l, li);
            }
        }
    }

    // ---- merge lane L (rows m<8 of each tile) with lane L+16 (rows m>=8) ----
    __syncthreads();   // done reading cand; safe to reuse LDS
#pragma unroll
    for (int j = 0; j < KSEL; ++j) {
        lval[tid * KSEL + j] = l[j];
        lidx[tid * KSEL + j] = li[j];
    }
    __syncthreads();

    if (!hi) {
        const float* a  = lval + (size_t)tid * KSEL;
        const int*   ai = lidx + (size_t)tid * KSEL;
        const float* b  = lval + (size_t)(tid + 16) * KSEL;
        const int*   bi = lidx + (size_t)(tid + 16) * KSEL;

        const size_t obase = ((size_t)batch * NPTS + q) * KSEL;
        int i = 0, j = 0;
#pragma unroll
        for (int t = 0; t < KSEL; ++t) {
            const float av = a[i], bv = b[j];
            const bool takeA = (av <= bv);
            const float v  = takeA ? av : bv;
            const int   id = takeA ? ai[i] : bi[j];
            i += takeA ? 1 : 0;
            j += takeA ? 0 : 1;
            outD[obase + t] = sqrtf(fmaxf(v, 0.0f));
            outI[obase + t] = id;
        }
    }
}

extern "C" void kernel_launch(void* const* d_in, const int* in_sizes, int n_in,
                              void* d_out, int out_size, void* d_ws, size_t ws_size,
                              hipStream_t stream) {
    (void)in_sizes; (void)n_in; (void)d_ws; (void)ws_size; (void)out_size;
    const float* src = (const float*)d_in[0];
    const float* dst = (const float*)d_in[1];
    float* outD = (float*)d_out;
    int*   outI = (int*)d_out + (size_t)4 * NPTS * KSEL;  // idx half of tuple output

    dim3 grid(4 * (NPTS / QPB));   // 256 blocks: 4 batches x 64
    dim3 block(THREADS);
    KNN_68152541053158_kernel<<<grid, block, 0, stream>>>(src, dst, outD, outI);
}